// LayerStacks_6262062317933
// MI455X (gfx1250) — compile-verified
//
#include <hip/hip_runtime.h>

typedef __attribute__((ext_vector_type(2))) float v2f;
typedef __attribute__((ext_vector_type(8))) float v8f;

#define NB      131072
#define L1D     1024      // 2*L1
#define NL2     16
#define NL3     32
#define NSTACK  8
#define MAXTILES ((NB / 16) + NSTACK)   // worst-case padded tile count = 8200
#define PF_DIST 512                     // prefetch distance in floats (2 KB)

// ---------------- bucketing kernels ----------------

__global__ void k_init(int* __restrict__ counts, int* __restrict__ cursor) {
  int i = threadIdx.x;
  if (i < NSTACK) { counts[i] = 0; cursor[i] = 0; }
}

__global__ void k_count(const int* __restrict__ idx, int* __restrict__ counts) {
  int i = blockIdx.x * blockDim.x + threadIdx.x;
  if (i < NB) atomicAdd(&counts[idx[i]], 1);
}

__global__ void k_scan(const int* __restrict__ counts, int* __restrict__ offs) {
  int o = 0;
  for (int b = 0; b < NSTACK; ++b) {
    offs[b] = o;
    o += (counts[b] + 15) & ~15;      // pad each bucket to multiple of 16
  }
  offs[NSTACK] = o;
}

__global__ void k_fillperm(int* __restrict__ perm) {
  int i = blockIdx.x * blockDim.x + threadIdx.x;
  if (i < NB + NSTACK * 16) perm[i] = -1;
}

__global__ void k_scatter(const int* __restrict__ idx, const int* __restrict__ offs,
                          int* __restrict__ cursor, int* __restrict__ perm) {
  int i = blockIdx.x * blockDim.x + threadIdx.x;
  if (i < NB) {
    int s = idx[i];
    int pos = atomicAdd(&cursor[s], 1);
    perm[offs[s] + pos] = i;
  }
}

// ---------------- main fused MLP kernel ----------------

__device__ __forceinline__ float shift_clamp(float v) {
  v = truncf(v);
  v = floorf(v * (1.0f / 64.0f));
  return fminf(fmaxf(v, 0.0f), 127.0f);   // relu subsumed by clamp at 0
}

__global__ __launch_bounds__(256) void k_main(
    const float* __restrict__ x,  const float* __restrict__ w1, const float* __restrict__ b1,
    const float* __restrict__ w2, const float* __restrict__ b2,
    const float* __restrict__ w_out, const float* __restrict__ b_out,
    const int* __restrict__ perm, const int* __restrict__ offs,
    float* __restrict__ out)
{
  // per-wave LDS staging: l1 tile 16x16 (stride 18), l2 tile 16x32 (stride 34)
  __shared__ float lds[8 * (16 * 18 + 16 * 34)];
  const int tid  = threadIdx.x;
  const int wave = tid >> 5;
  const int lane = tid & 31;
  const int m    = lane & 15;   // M (A-frag) / N (B,C,D frags)
  const int hi   = lane >> 4;   // K-half selector for 16x16x4 f32 frags
  float* l1s = &lds[wave * (16 * 18 + 16 * 34)];
  float* l2s = l1s + 16 * 18;

  const int total = offs[NSTACK];
  const int t     = blockIdx.x * 8 + wave;   // one 16-row tile per wave
  const int base  = t * 16;
  if (base >= total) return;

  // bucket (= stack index) of this tile
  int s = 0;
#pragma unroll
  for (int b = 1; b < NSTACK; ++b) if (base >= offs[b]) s = b;

  const int  rowRaw = perm[base + m];
  const bool valid  = rowRaw >= 0;
  const int  row0   = perm[base];            // first slot of a tile is always valid
  const int  row    = valid ? rowRaw : row0;

  // ---- layer 1: [16 rows x 1024] @ [1024 x 16] via 256 chained WMMAs ----
  // A frag (16x4 f32): lanes 0-15 -> K=k0,k0+1 ; lanes 16-31 -> K=k0+2,k0+3
  const float* xr = x  + (size_t)row * L1D + hi * 2;
  const float* wr = w1 + (size_t)(s * NL2 + m) * L1D + hi * 2;

  v8f acc = {};
  for (int k0 = 0; k0 < L1D; k0 += 32) {
    // keep the x stream ~2 KB ahead of consumption (global_prefetch_b8);
    // speculative prefetch past the allocation is silently dropped.
    __builtin_prefetch(xr + k0 + PF_DIST, 0, 3);
#pragma unroll
    for (int kk = 0; kk < 32; kk += 4) {
      v2f a  = *(const v2f*)(xr + k0 + kk);
      v2f wf = *(const v2f*)(wr + k0 + kk);
      acc = __builtin_amdgcn_wmma_f32_16x16x4_f32(false, a, false, wf,
                                                  (short)0, acc, false, false);
    }
  }

  // D layout: VGPR r holds (M = r + 8*hi, N = m). Quantize + transpose via LDS.
  const float bias1 = b1[s * NL2 + m];
#pragma unroll
  for (int r = 0; r < 8; ++r)
    l1s[(r + 8 * hi) * 18 + m] = shift_clamp(acc[r] + bias1);
  asm volatile("s_wait_dscnt 0" ::: "memory");

  // ---- layer 2: [16 x 16] @ [16 x 32] as two N-halves, K=16 -> 8 WMMAs ----
  const float* w2a = w2 + (size_t)(s * NL3 + m)      * NL2 + hi * 2;
  const float* w2b = w2 + (size_t)(s * NL3 + 16 + m) * NL2 + hi * 2;
  v8f acc2a = {}, acc2b = {};
#pragma unroll
  for (int k0 = 0; k0 < NL2; k0 += 4) {
    v2f a  = *(const v2f*)(l1s + m * 18 + k0 + hi * 2);   // M=m, K=k0+hi*2..+1
    v2f wa = *(const v2f*)(w2a + k0);
    v2f wb = *(const v2f*)(w2b + k0);
    acc2a = __builtin_amdgcn_wmma_f32_16x16x4_f32(false, a, false, wa,
                                                  (short)0, acc2a, false, false);
    acc2b = __builtin_amdgcn_wmma_f32_16x16x4_f32(false, a, false, wb,
                                                  (short)0, acc2b, false, false);
  }

  const float b2a = b2[s * NL3 + m];
  const float b2b = b2[s * NL3 + 16 + m];
#pragma unroll
  for (int r = 0; r < 8; ++r) {
    l2s[(r + 8 * hi) * 34 + m]      = shift_clamp(acc2a[r] + b2a);
    l2s[(r + 8 * hi) * 34 + 16 + m] = shift_clamp(acc2b[r] + b2b);
  }
  asm volatile("s_wait_dscnt 0" ::: "memory");

  // ---- layer 3: per-row 32-length dot with w_out[s], masked scatter-store ----
  if (hi == 0 && valid) {
    float sum = b_out[s];
#pragma unroll
    for (int k = 0; k < NL3; ++k)
      sum += l2s[m * 34 + k] * w_out[s * NL3 + k];
    out[row] = sum;
  }
}

// ---------------- launch ----------------

extern "C" void kernel_launch(void* const* d_in, const int* in_sizes, int n_in,
                              void* d_out, int out_size, void* d_ws, size_t ws_size,
                              hipStream_t stream) {
  const float* x     = (const float*)d_in[0];
  const int*   idx   = (const int*)  d_in[1];
  const float* w1    = (const float*)d_in[2];
  const float* b1    = (const float*)d_in[3];
  const float* w2    = (const float*)d_in[4];
  const float* b2    = (const float*)d_in[5];
  const float* w_out = (const float*)d_in[6];
  const float* b_out = (const float*)d_in[7];
  float* out = (float*)d_out;

  int* counts = (int*)d_ws;       // 8
  int* cursor = counts + 8;       // 8
  int* offs   = counts + 16;      // 9 (padded to 16)
  int* perm   = counts + 32;      // NB + 128

  k_init    <<<1, 32, 0, stream>>>(counts, cursor);
  k_count   <<<(NB + 255) / 256, 256, 0, stream>>>(idx, counts);
  k_scan    <<<1, 1, 0, stream>>>(counts, offs);
  k_fillperm<<<(NB + NSTACK * 16 + 255) / 256, 256, 0, stream>>>(perm);
  k_scatter <<<(NB + 255) / 256, 256, 0, stream>>>(idx, offs, cursor, perm);
  k_main    <<<(MAXTILES + 7) / 8, 256, 0, stream>>>(x, w1, b1, w2, b2,
                                                     w_out, b_out, perm, offs, out);
}